// GNODEModel_41214506172486
// MI455X (gfx1250) — compile-verified
//
#include <hip/hip_runtime.h>

// GNODE fused kernel for MI455X (gfx1250, wave32).
// f(y) = relu(y@W1+b1)@W2+b2 ; 10 RK4(3/8) steps, h=0.1; out = y@Wl+bl.
// Fully fused: state lives in registers, weights swizzled into LDS once.

#define NTOT   200000
#define CDIM   128
#define OUTC   64
#define STRIDE 132          // padded row stride (floats) in LDS stage -> conflict-free

typedef float v2f __attribute__((ext_vector_type(2)));
typedef float v8f __attribute__((ext_vector_type(8)));

__device__ __forceinline__ void lds_wait() {
  asm volatile("s_wait_dscnt 0" ::: "memory");
}

// D = A(16x4) x B(4x16) + C, f32.  A/B = v2f, C/D = v8f (CDNA5 WMMA layouts).
__device__ __forceinline__ v8f wmma4(v2f a, v2f b, v8f c) {
  return __builtin_amdgcn_wmma_f32_16x16x4_f32(false, a, false, b, (short)0, c,
                                               false, false);
}

// One 16xK (K=128) x Kx(16*TT) GEMM.  aptr: per-lane A base in LDS stage.
// wl: swizzled B-operand weights in LDS: [(kc*TT+t)*32+lane] v2f slots.
template <int TT>
__device__ __forceinline__ void gemm_tiles(const float* __restrict__ wl,
                                           int lane,
                                           const float* __restrict__ aptr,
                                           v8f* __restrict__ acc) {
#pragma unroll
  for (int t = 0; t < TT; ++t) acc[t] = (v8f)0.0f;
  const float* wlp = wl + lane * 2;
#pragma unroll 8
  for (int kc = 0; kc < 32; ++kc) {
    v2f a = *(const v2f*)(aptr + kc * 4);
#pragma unroll
    for (int t = 0; t < TT; ++t) {
      v2f b = *(const v2f*)(wlp + kc * TT * 64 + t * 64);
      acc[t] = wmma4(a, b, acc[t]);
    }
  }
}

__launch_bounds__(256, 1)
__global__ void gnode_kernel(const float* __restrict__ x,
                             const float* __restrict__ W1,
                             const float* __restrict__ b1,
                             const float* __restrict__ W2,
                             const float* __restrict__ b2,
                             const float* __restrict__ Wl,
                             const float* __restrict__ bl,
                             float* __restrict__ out) {
  // Swizzled weights: B-operand layout, one conflict-free ds_load_b64 per tile.
  __shared__ __attribute__((aligned(16))) float w1s[32 * 8 * 32 * 2];   // 64 KB
  __shared__ __attribute__((aligned(16))) float w2s[32 * 8 * 32 * 2];   // 64 KB
  __shared__ __attribute__((aligned(16))) float wls[32 * 4 * 32 * 2];   // 32 KB
  __shared__ __attribute__((aligned(16))) float stage[8 * 16 * STRIDE]; // ~66 KB

  const int tid   = threadIdx.x;
  const int lane  = tid & 31;
  const int wave  = tid >> 5;
  const int lrow  = lane & 15;
  const int half8 = (lane >> 4) << 3;   // 0 or 8

  // ---- fill swizzled weight LDS (once; weights stay L2-resident across blocks)
  for (int s = tid; s < 32 * 8 * 32; s += 256) {
    int ls = s & 31;
    int t  = (s >> 5) & 7;
    int kc = s >> 8;
    int k0 = kc * 4 + ((ls >> 4) << 1);
    int n  = t * 16 + (ls & 15);
    w1s[2 * s]     = W1[k0 * CDIM + n];
    w1s[2 * s + 1] = W1[(k0 + 1) * CDIM + n];
    w2s[2 * s]     = W2[k0 * CDIM + n];
    w2s[2 * s + 1] = W2[(k0 + 1) * CDIM + n];
  }
  for (int s = tid; s < 32 * 4 * 32; s += 256) {
    int ls = s & 31;
    int t  = (s >> 5) & 3;
    int kc = s >> 7;
    int k0 = kc * 4 + ((ls >> 4) << 1);
    int n  = t * 16 + (ls & 15);
    wls[2 * s]     = Wl[k0 * OUTC + n];
    wls[2 * s + 1] = Wl[(k0 + 1) * OUTC + n];
  }

  // ---- per-lane bias values (column n = 16t + lrow in C/D layout)
  float b1v[8], b2v[8], blv[4];
#pragma unroll
  for (int t = 0; t < 8; ++t) {
    b1v[t] = b1[t * 16 + lrow];
    b2v[t] = b2[t * 16 + lrow];
  }
#pragma unroll
  for (int t = 0; t < 4; ++t) blv[t] = bl[t * 16 + lrow];

  // ---- load x into state S (WMMA C/D layout), zero-fill tail rows
  const int rowbase = blockIdx.x * 128 + wave * 16;
  v8f S[8], K[8], acc[8];
#pragma unroll
  for (int t = 0; t < 8; ++t) {
#pragma unroll
    for (int r = 0; r < 8; ++r) {
      int row = rowbase + r + half8;
      S[t][r] = (row < NTOT) ? x[row * CDIM + t * 16 + lrow] : 0.0f;
    }
  }

  __syncthreads();  // swizzled weights visible

  float* st = stage + wave * 16 * STRIDE;          // this wave's private stage
  const float* aptr = st + lrow * STRIDE + ((lane >> 4) << 1);  // A-operand base

  // write a v8f-per-tile generator into the stage (row-major 16x128, padded)
  auto emit = [&](auto&& gen) {
#pragma unroll
    for (int t = 0; t < 8; ++t) {
      v8f e = gen(t);
#pragma unroll
      for (int r = 0; r < 8; ++r)
        st[(r + half8) * STRIDE + t * 16 + lrow] = e[r];
    }
    lds_wait();
  };

  // acc = f(stage) : GEMM1 -> +b1,relu -> stage -> GEMM2 -> +b2
  auto feval = [&]() {
    gemm_tiles<8>(w1s, lane, aptr, acc);
#pragma unroll
    for (int t = 0; t < 8; ++t) {
#pragma unroll
      for (int r = 0; r < 8; ++r)
        st[(r + half8) * STRIDE + t * 16 + lrow] = fmaxf(acc[t][r] + b1v[t], 0.0f);
    }
    lds_wait();
    gemm_tiles<8>(w2s, lane, aptr, acc);
#pragma unroll
    for (int t = 0; t < 8; ++t) acc[t] += b2v[t];
  };

  // RK4 3/8-rule, restructured so only S, K, acc are live:
  //   k1: S += h/8*k1;  K=k1;          a2 = S + 5h/24*K
  //   k2: S += 3h/8*k2;                a3 = S - 11h/24*K + 5h/8*k2
  //       K  = S + 7h/8*K - 11h/8*k2   (= a4 partial)
  //   k3: S += 3h/8*k3;                a4 = K + h*k3
  //   k4: S += h/8*k4
  const float hc    = 0.1f;
  const float c18   = hc * 0.125f;
  const float c38   = hc * 0.375f;
  const float c524  = hc * (5.0f / 24.0f);
  const float c1124 = hc * (11.0f / 24.0f);
  const float c58   = hc * 0.625f;
  const float c78   = hc * 0.875f;
  const float c118  = hc * 1.375f;

#pragma unroll 1
  for (int step = 0; step < 10; ++step) {
    emit([&](int t) { return S[t]; });                               // a1 = y
    feval();                                                         // k1
#pragma unroll
    for (int t = 0; t < 8; ++t) { K[t] = acc[t]; S[t] += c18 * acc[t]; }
    emit([&](int t) { return S[t] + c524 * K[t]; });                 // a2
    feval();                                                         // k2
#pragma unroll
    for (int t = 0; t < 8; ++t) S[t] += c38 * acc[t];
    emit([&](int t) { return S[t] - c1124 * K[t] + c58 * acc[t]; }); // a3
#pragma unroll
    for (int t = 0; t < 8; ++t) K[t] = S[t] + c78 * K[t] - c118 * acc[t];
    feval();                                                         // k3
#pragma unroll
    for (int t = 0; t < 8; ++t) S[t] += c38 * acc[t];
    emit([&](int t) { return K[t] + hc * acc[t]; });                 // a4
    feval();                                                         // k4
#pragma unroll
    for (int t = 0; t < 8; ++t) S[t] += c18 * acc[t];
  }

  // ---- final projection: out = y @ Wl + bl  (128 -> 64)
  emit([&](int t) { return S[t]; });
  gemm_tiles<4>(wls, lane, aptr, acc);
#pragma unroll
  for (int t = 0; t < 4; ++t) {
#pragma unroll
    for (int r = 0; r < 8; ++r) {
      int row = rowbase + r + half8;
      if (row < NTOT) out[row * OUTC + t * 16 + lrow] = acc[t][r] + blv[t];
    }
  }
}

extern "C" void kernel_launch(void* const* d_in, const int* in_sizes, int n_in,
                              void* d_out, int out_size, void* d_ws, size_t ws_size,
                              hipStream_t stream) {
  (void)in_sizes; (void)n_in; (void)out_size; (void)d_ws; (void)ws_size;
  const float* x  = (const float*)d_in[0];
  const float* W1 = (const float*)d_in[1];
  const float* b1 = (const float*)d_in[2];
  const float* W2 = (const float*)d_in[3];
  const float* b2 = (const float*)d_in[4];
  const float* Wl = (const float*)d_in[5];
  const float* bl = (const float*)d_in[6];
  float* out = (float*)d_out;

  dim3 grid((NTOT + 127) / 128);   // 1563 blocks x 256 threads (8 waves)
  gnode_kernel<<<grid, 256, 0, stream>>>(x, W1, b1, W2, b2, Wl, bl, out);
}